// FreqVEncoder_1657857376848
// MI455X (gfx1250) — compile-verified
//
#include <hip/hip_runtime.h>

// ---------------------------------------------------------------------------
// FreqVEncoder for MI455X (gfx1250, wave32)
//
// points: (65536,3) f32     freqs: (6) f32      cv: (96,16,32,32,32) f32
// out:    (65536, 96*16) f32
//
// Kernel 1 (encode): FP = P(16x3) x Fdiag(3x18) via V_WMMA_F32_16X16X4_F32,
//                    then sin/cos -> enc[n][pair=f*3+d][2] in workspace.
// Kernel 2 (gather): one block per point; threads 0..95 build per-volume
//                    trilinear weights/row-offsets in LDS; 256 threads do
//                    coalesced b64 corner-pair gathers + NT output stores.
// ---------------------------------------------------------------------------

typedef float v2f __attribute__((ext_vector_type(2)));
typedef float v8f __attribute__((ext_vector_type(8)));

#define N_POINTS 65536
#define NUM_COLS 1536          // 96 volumes * 16 channels
#define VOL_ELEMS 32768        // 32^3 per channel

// -------------------- Kernel 1: WMMA encode --------------------------------
// A (16x4 f32, wave32): lanes 0-15 hold K=0,1 in VGPR0,1; lanes 16-31 K=2,3.
// B (4x16 f32):         VGPR0: lanes0-15=K0 row, lanes16-31=K2 row;
//                       VGPR1: lanes0-15=K1 row, lanes16-31=K3 row.
// D (16x16 f32): lane l, VGPR r -> element (M = r + (l<16?0:8), N = l%16).
__global__ __launch_bounds__(256)
void fve_encode_wmma(const float* __restrict__ pts,
                     const float* __restrict__ freqs,
                     float* __restrict__ enc) {
  const int lane = threadIdx.x & 31;
  const int wave = threadIdx.x >> 5;
  const int tile = blockIdx.x * 8 + wave;      // 16 points per tile
  const int m    = lane & 15;
  const bool lo  = lane < 16;
  const int point = tile * 16 + m;

  const float px = pts[point * 3 + 0];
  const float py = pts[point * 3 + 1];
  const float pz = pts[point * 3 + 2];

  // A[M][K]: K0=px, K1=py, K2=pz, K3=0
  v2f a;
  a.x = lo ? px : pz;
  a.y = lo ? py : 0.0f;

  // B tile 1: column n -> pair n : (f = n/3, d = n%3), n=15 -> (f5,d0)
  const int pair = m;
  const int f    = pair / 3;
  const int d    = pair - f * 3;
  const float fv = freqs[f];
  v2f b1;
  b1.x = lo ? (d == 0 ? fv : 0.0f) : (d == 2 ? fv : 0.0f);  // K0 / K2 rows
  b1.y = lo ? (d == 1 ? fv : 0.0f) : 0.0f;                  // K1 / K3 rows

  v8f cz = {};
  v8f d1 = __builtin_amdgcn_wmma_f32_16x16x4_f32(
      false, a, false, b1, (short)0, cz, false, false);

  // B tile 2: cols 0,1 -> pairs 16,17 = (f5,d1),(f5,d2)
  const float f5 = freqs[5];
  v2f b2;
  b2.x = (!lo && m == 1) ? f5 : 0.0f;  // K2 row, d==2 at col 1
  b2.y = ( lo && m == 0) ? f5 : 0.0f;  // K1 row, d==1 at col 0
  v8f d2 = __builtin_amdgcn_wmma_f32_16x16x4_f32(
      false, a, false, b2, (short)0, cz, false, false);

  const int mbase = tile * 16 + (lo ? 0 : 8);
#pragma unroll
  for (int r = 0; r < 8; ++r) {
    float s, c;
    __sincosf(d1[r], &s, &c);
    float2* dst = (float2*)(enc + (size_t)(mbase + r) * 36 + (size_t)m * 2);
    *dst = make_float2(s, c);
  }
  if (m < 2) {
#pragma unroll
    for (int r = 0; r < 8; ++r) {
      float s, c;
      __sincosf(d2[r], &s, &c);
      float2* dst =
          (float2*)(enc + (size_t)(mbase + r) * 36 + (size_t)(16 + m) * 2);
      *dst = make_float2(s, c);
    }
  }
}

// -------------------- Kernel 2: trilinear gather ---------------------------
struct __attribute__((packed, aligned(4))) F2 { float x, y; };

__global__ __launch_bounds__(256)
void fve_gather(const float* __restrict__ cv,
                const float* __restrict__ enc,
                float* __restrict__ out) {
  __shared__ float4 sq[96 * 3];   // per volume: q0={wx0,wx1,noswap,0}, q1=wyz[4], q2=rows[4]
  const int n   = blockIdx.x;
  const int tid = threadIdx.x;

  if (tid < 96) {
    const int v  = tid;
    const int f  = v >> 4;           // v = f*16 + s*8 + k ; s duplicates coords
    const int k  = v & 7;
    const int bx = (k >> 2) & 1, by = (k >> 1) & 1, bz = k & 1;

    const float* e = enc + (size_t)n * 36 + f * 6;
    const float gx = e[0 + bx], gy = e[2 + by], gz = e[4 + bz];

    const float fx = fmaf(gx, 16.0f, 15.5f);   // ((x+1)*32 - 1)*0.5
    const float fy = fmaf(gy, 16.0f, 15.5f);
    const float fz = fmaf(gz, 16.0f, 15.5f);
    const float x0f = floorf(fx), y0f = floorf(fy), z0f = floorf(fz);
    const float wx = fx - x0f, wy = fy - y0f, wz = fz - z0f;
    const int x0 = (int)x0f, y0 = (int)y0f, z0 = (int)z0f;

    // x corner pair folded into one b64 load at xb (clamped), flipped on border
    const int xb = min(max(x0, 0), 30);
    float4 q0;
    q0.x = (x0 >= 0)  ? (1.0f - wx) : 0.0f;   // weight of x0 corner
    q0.y = (x0 <= 30) ? wx          : 0.0f;   // weight of x0+1 corner
    q0.z = (x0 == xb) ? 1.0f : 0.0f;          // noswap flag
    q0.w = 0.0f;

    const float wyv[2] = {1.0f - wy, wy};
    const float wzv[2] = {1.0f - wz, wz};
    float wq[4];
    int   rq[4];
#pragma unroll
    for (int j = 0; j < 4; ++j) {
      const int dy = j & 1, dz = j >> 1;
      const int yy = y0 + dy, zz = z0 + dz;
      const bool ok = (yy >= 0) & (yy <= 31) & (zz >= 0) & (zz <= 31);
      const int yc = min(max(yy, 0), 31), zc = min(max(zz, 0), 31);
      rq[j] = ((zc << 5) + yc) * 32 + xb;               // element offset
      wq[j] = ok ? wyv[dy] * wzv[dz] : 0.0f;
    }
    sq[v * 3 + 0] = q0;
    sq[v * 3 + 1] = make_float4(wq[0], wq[1], wq[2], wq[3]);
    sq[v * 3 + 2] = make_float4(__int_as_float(rq[0]), __int_as_float(rq[1]),
                                __int_as_float(rq[2]), __int_as_float(rq[3]));
  }
  __syncthreads();

  const size_t outbase = (size_t)n * NUM_COLS;
#pragma unroll
  for (int it = 0; it < 6; ++it) {
    const int c = it * 256 + tid;           // output column == v*16 + ch
    const int v = c >> 4;
    const float4 q0 = sq[v * 3 + 0];
    const float4 q1 = sq[v * 3 + 1];
    const float4 q2 = sq[v * 3 + 2];
    const float* base = cv + (size_t)c * VOL_ELEMS;
    const bool ns = (q0.z != 0.0f);
    const float wv[4] = {q1.x, q1.y, q1.z, q1.w};
    const int   rv[4] = {__float_as_int(q2.x), __float_as_int(q2.y),
                         __float_as_int(q2.z), __float_as_int(q2.w)};
    float acc = 0.0f;
#pragma unroll
    for (int j = 0; j < 4; ++j) {
      const F2 pr = *(const F2*)(base + rv[j]);   // x-corner pair, one b64
      const float v0 = ns ? pr.x : pr.y;
      const float v1 = ns ? pr.y : pr.x;
      acc = fmaf(wv[j], fmaf(q0.x, v0, q0.y * v1), acc);
    }
    __builtin_nontemporal_store(acc, out + outbase + c);  // stream past L2
  }
}

// ---------------------------------------------------------------------------
extern "C" void kernel_launch(void* const* d_in, const int* in_sizes, int n_in,
                              void* d_out, int out_size, void* d_ws,
                              size_t ws_size, hipStream_t stream) {
  const float* pts   = (const float*)d_in[0];   // (65536,3)
  const float* freqs = (const float*)d_in[1];   // (6)
  const float* cv    = (const float*)d_in[2];   // (96,16,32,32,32)
  float* out = (float*)d_out;
  float* enc = (float*)d_ws;                    // 65536*36 floats = 9.4 MB

  // 16 points per wave-tile, 8 tiles per 256-thread block -> 512 blocks
  fve_encode_wmma<<<N_POINTS / 128, 256, 0, stream>>>(pts, freqs, enc);
  // one block per point
  fve_gather<<<N_POINTS, 256, 0, stream>>>(cv, enc, out);
}